// GCN_8443905704050
// MI455X (gfx1250) — compile-verified
//
#include <hip/hip_runtime.h>

typedef __attribute__((ext_vector_type(2))) float v2f;
typedef __attribute__((ext_vector_type(8))) float v8f;

// ---------------------------------------------------------------------------
// Dense GEMM via V_WMMA_F32_16X16X4_F32 (full f32 precision, matches ref).
// One wave per 16x16 output tile. D = (RELU ? relu(A) : A) * W + bias.
// A: M x K row-major.  W: K x N row-major.  Dout: M x N row-major.
// K, N compile-time -> fully unrolled k-loop (32 wmma), immediate-offset
// global loads, no address SALU in the hot loop. M%16==0 (50000 = 3125*16),
// so every wave has EXEC all-1s and no divergence (WMMA requirement).
//
// f32 WMMA VGPR layouts (ISA 7.12.2):
//   A (16x4):  lanes 0-15 -> M=lane, v[0]=K0,v[1]=K1 ; lanes 16-31 -> K2,K3
//   B (4x16):  lane owns column n=lane&15; lane half selects K pair
//   C/D (16x16): acc[r] -> row (r + 8*(lane>>4)), col (lane&15)
// ---------------------------------------------------------------------------
template <int K, int N, bool RELU>
__global__ __launch_bounds__(32)
void gcn_gemm_wmma_f32(const float* __restrict__ A,
                       const float* __restrict__ W,
                       const float* __restrict__ bias,
                       float* __restrict__ Dout)
{
    const int lane  = threadIdx.x;        // 0..31
    const int hi    = lane >> 4;          // 0 or 1 (wave half)
    const int l16   = lane & 15;
    const int mtile = blockIdx.y << 4;
    const int ntile = blockIdx.x << 4;

    const int m    = mtile + l16;         // A row owned by this lane
    const int koff = hi << 1;             // 0 or 2

    // bias depends only on column -> same value in all 8 acc VGPRs
    const float bv = bias[ntile + l16];
    v8f acc;
#pragma unroll
    for (int r = 0; r < 8; ++r) acc[r] = bv;

    const float* __restrict__ arow = A + (size_t)m * K + koff;        // 8B-aligned
    const float* __restrict__ bcol = W + (size_t)koff * N + ntile + l16;

#pragma unroll
    for (int k0 = 0; k0 < K; k0 += 4) {
        v2f a = *(const v2f*)(arow + k0);                 // K = k0+koff, +1
        if (RELU) {
            a[0] = fmaxf(a[0], 0.0f);
            a[1] = fmaxf(a[1], 0.0f);
        }
        v2f b;
        b[0] = bcol[k0 * N];                              // K = k0+koff
        b[1] = bcol[(k0 + 1) * N];                        // K = k0+koff+1
        // (neg_a, A, neg_b, B, c_mod, C, reuse_a, reuse_b)
        acc = __builtin_amdgcn_wmma_f32_16x16x4_f32(
            false, a, false, b, (short)0, acc, false, false);
    }

#pragma unroll
    for (int r = 0; r < 8; ++r) {
        Dout[(size_t)(mtile + (hi << 3) + r) * N + ntile + l16] = acc[r];
    }
}

// ---------------------------------------------------------------------------
// SpMM scatter, 128 features: 32 lanes per edge, float4 (b128) gather from an
// L2-resident H (25.6 MB << 192 MB L2), 4 f32 atomic adds per lane into the
// segment-sum output. This stage dominates runtime (L2 atomic throughput).
// ---------------------------------------------------------------------------
__global__ __launch_bounds__(256)
void spmm_scatter128(const int* __restrict__ row, const int* __restrict__ col,
                     const float* __restrict__ val,
                     const float* __restrict__ H, float* __restrict__ out, int E)
{
    const long long t = (long long)blockIdx.x * blockDim.x + threadIdx.x;
    const int e = (int)(t >> 5);
    if (e >= E) return;
    const int f = ((int)t & 31) << 2;                    // 4 feats per lane
    const int r = row[e];
    const int c = col[e];
    const float v = val[e];
    const float4 h = *(const float4*)(H + (size_t)c * 128 + f);
    float* o = out + (size_t)r * 128 + f;
    atomicAdd(o + 0, v * h.x);
    atomicAdd(o + 1, v * h.y);
    atomicAdd(o + 2, v * h.z);
    atomicAdd(o + 3, v * h.w);
}

// ---------------------------------------------------------------------------
// SpMM scatter, 16 features: 16 lanes per edge, one atomic per lane.
// ---------------------------------------------------------------------------
__global__ __launch_bounds__(256)
void spmm_scatter16(const int* __restrict__ row, const int* __restrict__ col,
                    const float* __restrict__ val,
                    const float* __restrict__ H, float* __restrict__ out, int E)
{
    const long long t = (long long)blockIdx.x * blockDim.x + threadIdx.x;
    const int e = (int)(t >> 4);
    if (e >= E) return;
    const int f = (int)t & 15;
    const float g = val[e] * H[(size_t)col[e] * 16 + f];
    atomicAdd(out + (size_t)row[e] * 16 + f, g);
}

// ---------------------------------------------------------------------------
// Host-side orchestration (graph-capture safe: only async stream ops).
// Inputs: x, adj_row, adj_col, adj_val, W1, b1, W2, b2
// ---------------------------------------------------------------------------
extern "C" void kernel_launch(void* const* d_in, const int* in_sizes, int n_in,
                              void* d_out, int out_size, void* d_ws, size_t ws_size,
                              hipStream_t stream) {
    const float* x   = (const float*)d_in[0];
    const int*   row = (const int*)  d_in[1];
    const int*   col = (const int*)  d_in[2];
    const float* val = (const float*)d_in[3];
    const float* W1  = (const float*)d_in[4];
    const float* b1  = (const float*)d_in[5];
    const float* W2  = (const float*)d_in[6];
    const float* b2  = (const float*)d_in[7];
    float* out = (float*)d_out;

    constexpr int D_IN  = 128;
    constexpr int D_HID = 128;
    constexpr int N_CLS = 16;
    const int E = in_sizes[1];               // 800000 edges
    const int M = in_sizes[0] / D_IN;        // 50000 nodes (3125 * 16)

    // Workspace layout: H1 [M x D_HID] | AGG [M x D_HID] | H2 [M x N_CLS]
    char*  ws  = (char*)d_ws;
    float* H1  = (float*)ws;
    float* AGG = (float*)(ws + (size_t)M * D_HID * sizeof(float));
    float* H2  = (float*)(ws + 2 * (size_t)M * D_HID * sizeof(float));

    // Atomic accumulation targets must be zeroed every launch.
    hipMemsetAsync(AGG, 0, (size_t)M * D_HID * sizeof(float), stream);
    hipMemsetAsync(out, 0, (size_t)out_size * sizeof(float), stream);

    // Layer 1 dense: H1 = x @ W1 + b1     (8 x 3125 WMMA tiles)
    dim3 g1(D_HID / 16, M / 16);
    gcn_gemm_wmma_f32<D_IN, D_HID, false><<<g1, 32, 0, stream>>>(x, W1, b1, H1);

    // Layer 1 aggregate: AGG[r] += val * H1[c]   (32 lanes / edge)
    {
        const long long threads = (long long)E * 32;
        const int blk = 256;
        const unsigned nb = (unsigned)((threads + blk - 1) / blk);
        spmm_scatter128<<<nb, blk, 0, stream>>>(row, col, val, H1, AGG, E);
    }

    // Layer 2 dense with fused ReLU on input: H2 = relu(AGG) @ W2 + b2
    dim3 g2(N_CLS / 16, M / 16);
    gcn_gemm_wmma_f32<D_HID, N_CLS, true><<<g2, 32, 0, stream>>>(AGG, W2, b2, H2);

    // Layer 2 aggregate into d_out   (16 lanes / edge)
    {
        const long long threads = (long long)E * 16;
        const int blk = 256;
        const unsigned nb = (unsigned)((threads + blk - 1) / blk);
        spmm_scatter16<<<nb, blk, 0, stream>>>(row, col, val, H2, out, E);
    }
}